// GAT_5652176961662
// MI455X (gfx1250) — compile-verified
//
#include <hip/hip_runtime.h>

// ---------------------------------------------------------------------------
// GAT layer for MI455X (gfx1250, wave32).
//   h = x@W via V_WMMA_F32_16X16X4_F32 (exact f32 matrix path)
//   edge softmax via 3 atomic passes; node arrays live in the 192MB L2.
// ---------------------------------------------------------------------------

typedef __attribute__((ext_vector_type(2))) float v2f;
typedef __attribute__((ext_vector_type(8))) float v8f;

#define FDIM 128
#define HC 64          // HEADS * OUT_C
#define HEADS 8
#define OUTC 8
#define NEG_SLOPE 0.2f

__device__ __forceinline__ float leaky(float v) {
    return v > 0.0f ? v : NEG_SLOPE * v;
}

// float atomic max: signed-int max for non-negative, unsigned min for negative.
__device__ __forceinline__ void atomicMaxF(float* addr, float val) {
    if (val >= 0.0f) {
        atomicMax((int*)addr, __float_as_int(val));
    } else {
        atomicMin((unsigned int*)addr, __float_as_uint(val));
    }
}

// ---------------------------------------------------------------------------
// Kernel 1: h[N,64] = x[N,128] @ W[128,64] with f32 WMMA.
// grid = N/16 row tiles, block = 128 threads (4 waves). Wave w owns columns
// [16w, 16w+16). K loop: 32 steps of V_WMMA_F32_16X16X4_F32.
// A 16x4 layout: lane L holds row L%16; VGPR0/1 = K = (L/16)*2 + {0,1}.
// C/D 16x16 layout: VGPR v, lane L -> M = v + (L/16)*8, N = L%16.
// Store guard is a uniform branch: full tiles (always, since N%16==0 here)
// take the unguarded path -> no per-element exec-mask churn.
// ---------------------------------------------------------------------------
__global__ __launch_bounds__(128) void gat_gemm(const float* __restrict__ x,
                                                const float* __restrict__ W,
                                                float* __restrict__ h,
                                                int N) {
    const int lane    = threadIdx.x & 31;
    const int wave    = threadIdx.x >> 5;        // 0..3 -> 16-col group
    const int rowTile = blockIdx.x * 16;
    int row = rowTile + (lane & 15);
    if (row >= N) row = N - 1;                   // clamp loads, keep EXEC full
    const int colbase = wave * 16;
    const int col     = colbase + (lane & 15);
    const int koff    = (lane >> 4) * 2;         // 0 or 2

    const float* xrow = x + (size_t)row * FDIM;
    v8f c = {};
    #pragma unroll 4
    for (int k = 0; k < FDIM; k += 4) {
        const int kk = k + koff;
        float2 av = *(const float2*)(xrow + kk); // contiguous pair (8B aligned)
        v2f a, b;
        a.x = av.x;
        a.y = av.y;
        b.x = W[(size_t)kk * HC + col];
        b.y = W[(size_t)(kk + 1) * HC + col];
        // 8 args: (neg_a, A, neg_b, B, c_mod, C, reuse_a, reuse_b)
        c = __builtin_amdgcn_wmma_f32_16x16x4_f32(false, a, false, b,
                                                  (short)0, c, false, false);
    }

    const int mbase = (lane >> 4) * 8;           // 0 or 8
    const int n     = lane & 15;
    float* hp = h + (size_t)(rowTile + mbase) * HC + colbase + n;
    if (rowTile + 16 <= N) {                     // uniform: full tile
        #pragma unroll
        for (int v = 0; v < 8; ++v) {
            hp[(size_t)v * HC] = c[v];
        }
    } else {                                     // ragged last tile (unused here)
        #pragma unroll
        for (int v = 0; v < 8; ++v) {
            if (rowTile + mbase + v < N) hp[(size_t)v * HC] = c[v];
        }
    }
}

// ---------------------------------------------------------------------------
// Kernel 2: per (node, head): a_src/a_dst dot products; emax <- self-loop logit.
// ---------------------------------------------------------------------------
__global__ void gat_node_att(const float* __restrict__ h,
                             const float* __restrict__ att_src,
                             const float* __restrict__ att_dst,
                             float* __restrict__ a_src,
                             float* __restrict__ a_dst,
                             float* __restrict__ emax,
                             int NH) {
    const int t = blockIdx.x * blockDim.x + threadIdx.x;
    if (t >= NH) return;
    const int head = t & (HEADS - 1);
    const float* hp = h + (size_t)(t >> 3) * HC + head * OUTC;
    float as = 0.0f, ad = 0.0f;
    #pragma unroll
    for (int cidx = 0; cidx < OUTC; ++cidx) {
        const float hv = hp[cidx];
        as += hv * att_src[head * OUTC + cidx];
        ad += hv * att_dst[head * OUTC + cidx];
    }
    a_src[t] = as;
    a_dst[t] = ad;
    emax[t]  = leaky(as + ad);   // self-loop edge logit seeds the max
}

// ---------------------------------------------------------------------------
// Kernel 3: per (edge, head): atomic max of leaky logits into emax[dst].
// ---------------------------------------------------------------------------
__global__ void gat_edge_max(const long long* __restrict__ srci,
                             const long long* __restrict__ dsti,
                             const float* __restrict__ a_src,
                             const float* __restrict__ a_dst,
                             float* __restrict__ emax,
                             long long E) {
    const long long t = (long long)blockIdx.x * blockDim.x + threadIdx.x;
    if (t >= E * HEADS) return;
    const long long e = t >> 3;
    const int head    = (int)(t & (HEADS - 1));
    const long long s = srci[e];
    const long long d = dsti[e];
    const float logit = leaky(a_src[s * HEADS + head] + a_dst[d * HEADS + head]);
    atomicMaxF(&emax[d * HEADS + head], logit);
}

// ---------------------------------------------------------------------------
// Kernel 4: self-loop contribution seeds denom and numerator (num == d_out).
// ---------------------------------------------------------------------------
__global__ void gat_self_init(const float* __restrict__ h,
                              const float* __restrict__ a_src,
                              const float* __restrict__ a_dst,
                              const float* __restrict__ emax,
                              float* __restrict__ denom,
                              float* __restrict__ num,
                              int NH) {
    const int t = blockIdx.x * blockDim.x + threadIdx.x;
    if (t >= NH) return;
    const float eself = leaky(a_src[t] + a_dst[t]);
    const float exs   = __expf(eself - emax[t]);
    denom[t] = exs;
    const int head = t & (HEADS - 1);
    const float* hp = h + (size_t)(t >> 3) * HC + head * OUTC;
    float* np       = num + (size_t)(t >> 3) * HC + head * OUTC;
    #pragma unroll
    for (int cidx = 0; cidx < OUTC; ++cidx) np[cidx] = exs * hp[cidx];
}

// ---------------------------------------------------------------------------
// Kernel 5: main edge pass. 64 threads per edge (head = r/8, ch = r%8):
//   ex = exp(leaky(a_src[s]+a_dst[d]) - emax[d]);  num[d] += ex*h[s];
//   denom[d] += ex (one lane per head).
// Gathers/atomics hit L2 (all node arrays < 192MB); edge stream from HBM.
// ---------------------------------------------------------------------------
__global__ void gat_edge_acc(const long long* __restrict__ srci,
                             const long long* __restrict__ dsti,
                             const float* __restrict__ a_src,
                             const float* __restrict__ a_dst,
                             const float* __restrict__ emax,
                             const float* __restrict__ h,
                             float* __restrict__ denom,
                             float* __restrict__ num,
                             long long E) {
    const long long t = (long long)blockIdx.x * blockDim.x + threadIdx.x;
    if (t >= E * (long long)HC) return;
    const long long e = t >> 6;
    const int r       = (int)(t & (HC - 1));
    const int head    = r >> 3;
    const int ch      = r & (OUTC - 1);
    const long long s = srci[e];
    const long long d = dsti[e];
    const float logit = leaky(a_src[s * HEADS + head] + a_dst[d * HEADS + head]);
    const float ex    = __expf(logit - emax[d * HEADS + head]);
    atomicAdd(&num[d * HC + r], ex * h[s * HC + r]);
    if (ch == 0) atomicAdd(&denom[d * HEADS + head], ex);
}

// ---------------------------------------------------------------------------
// Kernel 6: out = num / denom + bias   (in place on d_out)
// ---------------------------------------------------------------------------
__global__ void gat_final(const float* __restrict__ denom,
                          const float* __restrict__ bias,
                          float* __restrict__ out,
                          long long total) {
    const long long t = (long long)blockIdx.x * blockDim.x + threadIdx.x;
    if (t >= total) return;
    const int r = (int)(t & (HC - 1));
    out[t] = out[t] / denom[(t >> 6) * HEADS + (r >> 3)] + bias[r];
}

extern "C" void kernel_launch(void* const* d_in, const int* in_sizes, int n_in,
                              void* d_out, int out_size, void* d_ws, size_t ws_size,
                              hipStream_t stream) {
    const float* x          = (const float*)d_in[0];
    const long long* ei     = (const long long*)d_in[1];  // int64 [2,E]
    const float* W          = (const float*)d_in[2];
    const float* att_src    = (const float*)d_in[3];
    const float* att_dst    = (const float*)d_in[4];
    const float* bias       = (const float*)d_in[5];

    const int N       = in_sizes[0] / FDIM;
    const long long E = (long long)in_sizes[1] / 2;
    const long long* srci = ei;
    const long long* dsti = ei + E;

    // Workspace carve-out: N*(64 + 4*8) floats = 38.4 MB for N=100k.
    float* ws     = (float*)d_ws;
    float* h      = ws;                                   // N*64
    float* a_src  = h + (size_t)N * HC;                   // N*8
    float* a_dst  = a_src + (size_t)N * HEADS;            // N*8
    float* emax   = a_dst + (size_t)N * HEADS;            // N*8
    float* denom  = emax + (size_t)N * HEADS;             // N*8
    float* out    = (float*)d_out;                        // also numerator accum

    const int tiles = (N + 15) / 16;
    gat_gemm<<<tiles, 128, 0, stream>>>(x, W, h, N);

    const int NH = N * HEADS;
    gat_node_att<<<(NH + 255) / 256, 256, 0, stream>>>(h, att_src, att_dst,
                                                       a_src, a_dst, emax, NH);

    const long long tmax = E * HEADS;
    gat_edge_max<<<(int)((tmax + 255) / 256), 256, 0, stream>>>(srci, dsti,
                                                                a_src, a_dst,
                                                                emax, E);

    gat_self_init<<<(NH + 255) / 256, 256, 0, stream>>>(h, a_src, a_dst, emax,
                                                        denom, out, NH);

    const long long tacc = E * (long long)HC;
    gat_edge_acc<<<(int)((tacc + 255) / 256), 256, 0, stream>>>(srci, dsti,
                                                                a_src, a_dst,
                                                                emax, h,
                                                                denom, out, E);

    const long long tout = (long long)N * HC;
    gat_final<<<(int)((tout + 255) / 256), 256, 0, stream>>>(denom, bias, out,
                                                             tout);
}